// StackedLSTMModel_3917010174478
// MI455X (gfx1250) — compile-verified
//
#include <hip/hip_runtime.h>

#define B_    64
#define T_    2048
#define D_    256
#define H4_   1024
#define NCLS_ 27
#define KC2_  32          // D_/8 : K handled in chunks of 8 (2 WMMA k-steps)
#define NT_   64          // H4_/16 : number of 16-wide column tiles

// LDS row strides (floats) chosen for bank-conflict-free access (64 banks)
#define CSTR_ 260   // 260  % 64 == 4
#define ZSTR_ 1028  // 1028 % 64 == 4

typedef __attribute__((ext_vector_type(2))) float v2f;
typedef __attribute__((ext_vector_type(4))) float v4f;
typedef __attribute__((ext_vector_type(8))) float v8f;

__device__ __forceinline__ v8f wmma_f32(v2f a, v2f b, v8f c) {
  // V_WMMA_F32_16X16X4_F32 : D = A(16x4) * B(4x16) + C(16x16), full f32
  return __builtin_amdgcn_wmma_f32_16x16x4_f32(false, a, false, b, (short)0, c,
                                               false, false);
}

// gfx1250 hardware transcendental: V_TANH_F32
__device__ __forceinline__ float htanh(float x) {
  return __builtin_amdgcn_tanhf(x);
}
// sigmoid(x) == 0.5*tanh(x/2) + 0.5  -> single TRANS op + FMA
__device__ __forceinline__ float sigm(float x) {
  return __builtin_fmaf(0.5f, htanh(0.5f * x), 0.5f);
}

// ---------------------------------------------------------------------------
// Kernel W: repack Wh[256][1024] into B-fragment-major layout:
//   pWh[nt][kc2][lane][4]  (lane's b128 = B-frags for kc=2*kc2 and 2*kc2+1)
// ---------------------------------------------------------------------------
__global__ __launch_bounds__(256)
void pack_wh_kernel(const float* __restrict__ Wh, float* __restrict__ pWh) {
  const int idx  = blockIdx.x * 256 + threadIdx.x;     // 0..65535
  const int lane = idx & 31;
  const int kc2  = (idx >> 5) & 31;
  const int nt   = idx >> 10;
  const int half = lane >> 4, l16 = lane & 15;
  const int n    = nt * 16 + l16;
  const int k0   = kc2 * 8 + 2 * half;
  v4f v;
  v.x = Wh[(long)(k0 + 0) * H4_ + n];
  v.y = Wh[(long)(k0 + 1) * H4_ + n];
  v.z = Wh[(long)(k0 + 4) * H4_ + n];
  v.w = Wh[(long)(k0 + 5) * H4_ + n];
  *(v4f*)(pWh + (long)idx * 4) = v;
}

// ---------------------------------------------------------------------------
// Kernel A: zx = embed[tokens] @ Wi + bh, stored FRAG-PACKED for the scan:
//   zxp[t][blk][nt][lane][8]  (lane's C-frag dump, 1KB contiguous per tile)
// M-tile = 16 batch rows of one block at fixed t (matches scan acc tile).
// grid = (T_*4, 8), block = 256 (8 waves, one n-tile each).
// ---------------------------------------------------------------------------
__global__ __launch_bounds__(256)
void zx_gemm_kernel(const int* __restrict__ tokens,
                    const float* __restrict__ embed,
                    const float* __restrict__ Wi,
                    const float* __restrict__ bh,
                    float* __restrict__ zxp) {
  const int lane = threadIdx.x & 31;
  const int wave = threadIdx.x >> 5;
  const int half = lane >> 4;
  const int l16  = lane & 15;

  const int t   = blockIdx.x >> 2;
  const int blk = blockIdx.x & 3;
  const int nt  = blockIdx.y * 8 + wave;
  const int n0  = nt * 16;

  // Lane owns batch row blk*16 + l16 at time t.
  const int tok = tokens[(blk * 16 + l16) * T_ + t];
  const float* arow = embed + (long)tok * D_;

  v8f acc = {};
#pragma unroll 4
  for (int kc = 0; kc < D_ / 4; ++kc) {
    const int k0 = kc * 4;
    v2f a = *(const v2f*)(arow + k0 + 2 * half);   // A frag (b64)
    v2f b;
    b.x = Wi[(long)(k0 + 2 * half) * H4_ + n0 + l16];
    b.y = Wi[(long)(k0 + 2 * half + 1) * H4_ + n0 + l16];
    acc = wmma_f32(a, b, acc);
  }
  const float bias = bh[n0 + l16];
#pragma unroll
  for (int j = 0; j < 8; ++j) acc[j] += bias;

  float* dst = zxp + ((((long)t * 4 + blk) * NT_ + nt) * 32 + lane) * 8;
  v4f lo = {acc[0], acc[1], acc[2], acc[3]};
  v4f hi = {acc[4], acc[5], acc[6], acc[7]};
  *(v4f*)(dst)     = lo;
  *(v4f*)(dst + 4) = hi;
}

// ---------------------------------------------------------------------------
// Kernel B: sequential LSTM scan. 4 workgroups (16 batch rows each),
// 1024 threads = 32 waves; wave w owns column tiles 2w, 2w+1.
// h lives in LDS in A-fragment layout hf[kc2][lane][4] (b128 in/out),
// c in padded row layout, z in padded row layout.
// ---------------------------------------------------------------------------
template <bool FUSE_WI>
__global__ __launch_bounds__(1024)
void lstm_scan_kernel(const int* __restrict__ tokens,
                      const float* __restrict__ embed,
                      const float* __restrict__ Wi,
                      const float* __restrict__ Wh,
                      const float* __restrict__ bh,
                      const float* __restrict__ Wo,
                      const float* __restrict__ bo,
                      const float* __restrict__ zxp,
                      const float* __restrict__ pWh,
                      float* __restrict__ out) {
  extern __shared__ float smem[];
  float* hf    = smem;                          // [KC2_*32*4] = 4096 floats
  float* c_buf = smem + KC2_ * 32 * 4;          // [16][CSTR_]
  float* z_buf = smem + KC2_ * 32 * 4 + 16 * CSTR_;  // [16][ZSTR_]

  const int blk  = blockIdx.x;                  // batch rows [blk*16, +16)
  const int tid  = threadIdx.x;
  const int lane = tid & 31;
  const int wave = tid >> 5;                    // 0..31
  const int half = lane >> 4;
  const int l16  = lane & 15;

  for (int i = tid; i < KC2_ * 32 * 4; i += 1024) hf[i] = 0.0f;
  for (int i = tid; i < 16 * CSTR_;    i += 1024) c_buf[i] = 0.0f;
  __syncthreads();

  const int nta = 2 * wave, ntb = 2 * wave + 1;
  const int n0a = nta * 16, n0b = ntb * 16;
  const float bias_a = FUSE_WI ? bh[n0a + l16] : 0.0f;
  const float bias_b = FUSE_WI ? bh[n0b + l16] : 0.0f;

  // gate-phase mapping: this thread owns h[row gr][cols gc0,gc0+1,gc0+4,gc0+5]
  // == exactly the 4 floats of hf word-group (wave*32+lane)*4  (one b128)
  const int gr  = l16;
  const int gc0 = 8 * wave + 2 * half;

  for (int t = 0; t < T_; ++t) {
    v8f accA, accB;

    if constexpr (!FUSE_WI) {
      // acc init = precomputed zx frags (bias folded in); 4 coalesced b128
      const float* src = zxp + (((long)t * 4 + blk) * NT_) * 32 * 8;
      const float* pa  = src + ((long)nta * 32 + lane) * 8;
      const float* pb  = src + ((long)ntb * 32 + lane) * 8;
      v4f a0 = *(const v4f*)pa, a1 = *(const v4f*)(pa + 4);
      v4f b0 = *(const v4f*)pb, b1 = *(const v4f*)(pb + 4);
      accA[0] = a0.x; accA[1] = a0.y; accA[2] = a0.z; accA[3] = a0.w;
      accA[4] = a1.x; accA[5] = a1.y; accA[6] = a1.z; accA[7] = a1.w;
      accB[0] = b0.x; accB[1] = b0.y; accB[2] = b0.z; accB[3] = b0.w;
      accB[4] = b1.x; accB[5] = b1.y; accB[6] = b1.z; accB[7] = b1.w;
      if (t + 1 < T_) {
        __builtin_prefetch(pa + (long)4 * NT_ * 32 * 8, 0, 1);
        __builtin_prefetch(pb + (long)4 * NT_ * 32 * 8, 0, 1);
      }
    } else {
#pragma unroll
      for (int j = 0; j < 8; ++j) { accA[j] = bias_a; accB[j] = bias_b; }
      // accumulate x_t @ Wi on the fly
      const int tokr = tokens[(blk * 16 + l16) * T_ + t];
      const float* xrow = embed + (long)tokr * D_;
#pragma unroll 4
      for (int kc2 = 0; kc2 < KC2_; ++kc2) {
        const int c0 = kc2 * 8 + 2 * half;
        v2f a01 = *(const v2f*)(xrow + c0);
        v2f a23 = *(const v2f*)(xrow + c0 + 4);
        v2f b;
        b.x = Wi[(long)(c0 + 0) * H4_ + n0a + l16];
        b.y = Wi[(long)(c0 + 1) * H4_ + n0a + l16];
        accA = wmma_f32(a01, b, accA);
        b.x = Wi[(long)(c0 + 4) * H4_ + n0a + l16];
        b.y = Wi[(long)(c0 + 5) * H4_ + n0a + l16];
        accA = wmma_f32(a23, b, accA);
        b.x = Wi[(long)(c0 + 0) * H4_ + n0b + l16];
        b.y = Wi[(long)(c0 + 1) * H4_ + n0b + l16];
        accB = wmma_f32(a01, b, accB);
        b.x = Wi[(long)(c0 + 4) * H4_ + n0b + l16];
        b.y = Wi[(long)(c0 + 5) * H4_ + n0b + l16];
        accB = wmma_f32(a23, b, accB);
      }
    }

    // z += h @ Wh : A from LDS (b128, shared by both tiles), B packed (b128)
#pragma unroll 4
    for (int kc2 = 0; kc2 < KC2_; ++kc2) {
      v4f a = *(const v4f*)(hf + (kc2 * 32 + lane) * 4);
      v2f a01 = {a.x, a.y};
      v2f a23 = {a.z, a.w};
      if constexpr (!FUSE_WI) {
        v4f ba = *(const v4f*)(pWh + ((long)(nta * 32 + kc2) * 32 + lane) * 4);
        v4f bb = *(const v4f*)(pWh + ((long)(ntb * 32 + kc2) * 32 + lane) * 4);
        v2f b01a = {ba.x, ba.y}, b23a = {ba.z, ba.w};
        v2f b01b = {bb.x, bb.y}, b23b = {bb.z, bb.w};
        accA = wmma_f32(a01, b01a, accA);
        accA = wmma_f32(a23, b23a, accA);
        accB = wmma_f32(a01, b01b, accB);
        accB = wmma_f32(a23, b23b, accB);
      } else {
        const int c0 = kc2 * 8 + 2 * half;
        v2f b;
        b.x = Wh[(long)(c0 + 0) * H4_ + n0a + l16];
        b.y = Wh[(long)(c0 + 1) * H4_ + n0a + l16];
        accA = wmma_f32(a01, b, accA);
        b.x = Wh[(long)(c0 + 4) * H4_ + n0a + l16];
        b.y = Wh[(long)(c0 + 5) * H4_ + n0a + l16];
        accA = wmma_f32(a23, b, accA);
        b.x = Wh[(long)(c0 + 0) * H4_ + n0b + l16];
        b.y = Wh[(long)(c0 + 1) * H4_ + n0b + l16];
        accB = wmma_f32(a01, b, accB);
        b.x = Wh[(long)(c0 + 4) * H4_ + n0b + l16];
        b.y = Wh[(long)(c0 + 5) * H4_ + n0b + l16];
        accB = wmma_f32(a23, b, accB);
      }
    }

    // scatter z tiles to LDS (bank-conflict-free via ZSTR_ pad)
#pragma unroll
    for (int j = 0; j < 8; ++j) {
      const int r = j + half * 8;
      z_buf[r * ZSTR_ + n0a + l16] = accA[j];
      z_buf[r * ZSTR_ + n0b + l16] = accB[j];
    }
    __syncthreads();

    // gates; each thread produces one b128 A-frag chunk of the new h.
    // 5 hardware V_TANH_F32 per element-group of work; short serial tail.
    const int cols[4] = {gc0, gc0 + 1, gc0 + 4, gc0 + 5};
    float hvv[4];
#pragma unroll
    for (int e = 0; e < 4; ++e) {
      const int c = cols[e];
      const float zi = z_buf[gr * ZSTR_ + c];
      const float zf = z_buf[gr * ZSTR_ + c + 256];
      const float zg = z_buf[gr * ZSTR_ + c + 512];
      const float zo = z_buf[gr * ZSTR_ + c + 768];
      float cc = sigm(zf) * c_buf[gr * CSTR_ + c] + sigm(zi) * htanh(zg);
      hvv[e] = sigm(zo) * htanh(cc);
      c_buf[gr * CSTR_ + c] = cc;
    }
    v4f hv = {hvv[0], hvv[1], hvv[2], hvv[3]};
    *(v4f*)(hf + (wave * 32 + lane) * 4) = hv;   // ds_store_b128
    __syncthreads();
  }

  // out = h @ Wo + bo  (16 x 27 per block; read h back out of frag layout)
  for (int i = tid; i < 16 * NCLS_; i += 1024) {
    const int r = i / NCLS_, cls = i % NCLS_;
    float s = bo[cls];
    for (int k = 0; k < D_; ++k) {
      const int kc2 = k >> 3, rem = k & 7;
      const int hh  = (rem >> 1) & 1;
      const int e   = (rem & 1) + 2 * (rem >> 2);
      s += hf[(kc2 * 32 + r + 16 * hh) * 4 + e] * Wo[k * NCLS_ + cls];
    }
    out[(blk * 16 + r) * NCLS_ + cls] = s;
  }
}

extern "C" void kernel_launch(void* const* d_in, const int* in_sizes, int n_in,
                              void* d_out, int out_size, void* d_ws, size_t ws_size,
                              hipStream_t stream) {
  const int*   tokens = (const int*)d_in[0];
  // d_in[1], d_in[2]: node spans -- dead code in the reference
  const float* embed  = (const float*)d_in[3];
  const float* Wi     = (const float*)d_in[4];
  const float* Wh     = (const float*)d_in[5];
  const float* bh     = (const float*)d_in[6];
  const float* Wo     = (const float*)d_in[7];
  const float* bo     = (const float*)d_in[8];
  float*       out    = (float*)d_out;

  const size_t zx_bytes  = (size_t)B_ * T_ * H4_ * sizeof(float);       // 512 MiB
  const size_t pwh_bytes = (size_t)NT_ * KC2_ * 32 * 4 * sizeof(float); // 1 MiB
  const size_t lds_bytes =
      (size_t)(KC2_ * 32 * 4 + 16 * CSTR_ + 16 * ZSTR_) * sizeof(float);

  if (ws_size >= zx_bytes + pwh_bytes) {
    float* zxp = (float*)d_ws;
    float* pWh = (float*)((char*)d_ws + zx_bytes);
    pack_wh_kernel<<<NT_ * KC2_ * 32 / 256, 256, 0, stream>>>(Wh, pWh);
    zx_gemm_kernel<<<dim3(T_ * 4, NT_ / 8), 256, 0, stream>>>(tokens, embed, Wi,
                                                              bh, zxp);
    lstm_scan_kernel<false><<<B_ / 16, 1024, lds_bytes, stream>>>(
        tokens, embed, Wi, Wh, bh, Wo, bo, zxp, pWh, out);
  } else {
    lstm_scan_kernel<true><<<B_ / 16, 1024, lds_bytes, stream>>>(
        tokens, embed, Wi, Wh, bh, Wo, bo, nullptr, nullptr, out);
  }
}